// L1neighs_Aggregator_20375324852396
// MI455X (gfx1250) — compile-verified
//
#include <hip/hip_runtime.h>
#include <cstdint>
#include <cstddef>

#define WAVES_PER_BLOCK 8
#define BLOCK_SIZE (WAVES_PER_BLOCK * 32)
#define EMBED_DIM 128
#define MAX_DEG 32
#define PIPE_DEPTH 4   // async rows in flight per wave

// CDNA5 async global->LDS copy path (ASYNCcnt-tracked), if the toolchain
// exposes the gfx1250 builtins. Falls back to direct b128 vector gathers.
#if defined(__has_builtin)
#if __has_builtin(__builtin_amdgcn_global_load_async_to_lds_b128) && \
    __has_builtin(__builtin_amdgcn_s_wait_asynccnt)
#define USE_ASYNC_LDS 1
#endif
#endif

// Builtin signature (from hipcc diagnostic):
//   void __builtin_amdgcn_global_load_async_to_lds_b128(
//       int __attribute__((vector_size(16))) AS(1)* gsrc,
//       int __attribute__((vector_size(16))) AS(3)* ldst,
//       imm int offset, imm int cpol)
typedef int vi4 __attribute__((vector_size(16)));
typedef __attribute__((address_space(1))) vi4 gbl_vi4_t;
typedef __attribute__((address_space(3))) vi4 lds_vi4_t;

__global__ __launch_bounds__(BLOCK_SIZE)
void l1_mean_pool_kernel(const int* __restrict__ neighbors,
                         const int* __restrict__ lengths,
                         const float* __restrict__ a2e,
                         float* __restrict__ out,
                         int n_nodes) {
    const int lane = threadIdx.x & 31;   // wave32
    const int wave = threadIdx.x >> 5;
    const int node = blockIdx.x * WAVES_PER_BLOCK + wave;
    if (node >= n_nodes) return;

    // One vector load grabs all 32 neighbor indices for this wave's node;
    // per-iteration index comes back as a scalar via v_readlane.
    const int nb_lane = neighbors[node * MAX_DEG + lane];
    const int len = __builtin_amdgcn_readfirstlane(lengths[node]);

    float4 acc = make_float4(0.0f, 0.0f, 0.0f, 0.0f);

#if defined(USE_ASYNC_LDS)
    __shared__ float stage[WAVES_PER_BLOCK][PIPE_DEPTH][EMBED_DIM];

    // per-lane source pointer for neighbor row g
    auto row_src = [&](int g) -> const float* {
        const int nb = __builtin_amdgcn_readlane(nb_lane, g);
        return a2e + (size_t)nb * EMBED_DIM + lane * 4;
    };
    // per-lane LDS staging address for slot s
    auto slot_ptr = [&](int s) -> float* {
        return &stage[wave][s][lane * 4];
    };
    auto consume = [&](int g) {
        const float4 v = *(const float4*)slot_ptr(g & (PIPE_DEPTH - 1));
        acc.x += v.x; acc.y += v.y; acc.z += v.z; acc.w += v.w;
    };

    // prologue: fill the pipe (up to PIPE_DEPTH rows in flight)
    for (int i = 0; i < len && i < PIPE_DEPTH; ++i) {
        __builtin_amdgcn_global_load_async_to_lds_b128(
            (gbl_vi4_t*)row_src(i), (lds_vi4_t*)slot_ptr(i & (PIPE_DEPTH - 1)), 0, 0);
    }

    int g = 0;
    // steady state: 4 rows outstanding; asynccnt<=3 => oldest (row g) landed
    for (; g + PIPE_DEPTH < len; ++g) {
        if (g + 2 * PIPE_DEPTH < len) {   // warm L2 beyond the async window
            const int nbp = __builtin_amdgcn_readlane(nb_lane, g + 2 * PIPE_DEPTH);
            __builtin_prefetch(a2e + (size_t)nbp * EMBED_DIM + lane * 4, 0, 3);
        }
        __builtin_amdgcn_s_wait_asynccnt(PIPE_DEPTH - 1);
        consume(g);
        // fence: our ds_load must drain before the async engine overwrites
        // this slot (reused by the issue below) — WAR guard
        asm volatile("s_wait_dscnt 0" ::: "memory");
        __builtin_amdgcn_global_load_async_to_lds_b128(
            (gbl_vi4_t*)row_src(g + PIPE_DEPTH),
            (lds_vi4_t*)slot_ptr((g + PIPE_DEPTH) & (PIPE_DEPTH - 1)), 0, 0);
    }
    // tail: <=PIPE_DEPTH rows left in flight; drain them all
    if (g < len) {
        __builtin_amdgcn_s_wait_asynccnt(0);
        for (; g < len; ++g) consume(g);
    }
#else
    for (int g = 0; g < len; ++g) {
        const int nb = __builtin_amdgcn_readlane(nb_lane, g);
        if (g + 4 < len) {
            const int nbp = __builtin_amdgcn_readlane(nb_lane, g + 4);
            __builtin_prefetch(a2e + (size_t)nbp * EMBED_DIM + lane * 4, 0, 3);
        }
        const float4 v =
            ((const float4*)(a2e + (size_t)nb * EMBED_DIM))[lane];
        acc.x += v.x; acc.y += v.y; acc.z += v.z; acc.w += v.w;
    }
#endif

    const float inv = (len > 0) ? (1.0f / (float)len) : 0.0f;
    const float4 r = make_float4(acc.x * inv, acc.y * inv, acc.z * inv, acc.w * inv);
    ((float4*)(out + (size_t)node * EMBED_DIM))[lane] = r;
}

extern "C" void kernel_launch(void* const* d_in, const int* in_sizes, int n_in,
                              void* d_out, int out_size, void* d_ws, size_t ws_size,
                              hipStream_t stream) {
    // setup_inputs order: node(unused), neighbors, lengths, a2e
    const int*   neighbors = (const int*)d_in[1];
    const int*   lengths   = (const int*)d_in[2];
    const float* a2e       = (const float*)d_in[3];
    float*       out       = (float*)d_out;
    const int n_nodes = in_sizes[2];   // one length per node

    dim3 block(BLOCK_SIZE);
    dim3 grid((n_nodes + WAVES_PER_BLOCK - 1) / WAVES_PER_BLOCK);
    hipLaunchKernelGGL(l1_mean_pool_kernel, grid, block, 0, stream,
                       neighbors, lengths, a2e, out, n_nodes);
}